// SubnetworkBlock_4312147165697
// MI455X (gfx1250) — compile-verified
//
#include <hip/hip_runtime.h>
#include <math.h>

// ---------------------------------------------------------------------------
// SubnetworkBlock for MI455X (gfx1250, wave32).
//
// Structure:
//   k1: fused forward + analytic d2(out)/dx2.  Layer-2 matmuls (K=10,N=6,
//       shared W2 across h / h' / h'') run on V_WMMA_F32_16X16X4_F32:
//         A  = W2^T zero-padded to 16x12 (per-subnet constant, 6 VGPRs)
//         B  = 16 samples' layer-1 vectors; lanes 16..31 duplicate samples
//              0..15 so each B VGPR is built with one v_cndmask (no shuffles)
//         D  = v / v' / v'' with unit n in VGPR n, sample m in lane m.
//       Writes raw out straight into d_out[b*S+s] (L2 merges the stride-S
//       stores; 192MB L2 >> 33MB output) and atomic-adds per-subnet
//       {sum(out), sum(out^2), sum(g2^2)}.
//   k2: per-subnet mean / clamped std / loss reduction (1 block).
//   k3: in-place vectorized (out-mean)*rstd normalize, float4.
// ---------------------------------------------------------------------------

typedef float v2f __attribute__((ext_vector_type(2)));
typedef float v8f __attribute__((ext_vector_type(8)));

#define SN    256      // subnets
#define BN    32768    // batch
#define NH1   10
#define NH2   6
#define CHUNKS 16      // batch chunks (grid.y of k1)
#define WAVES  8       // 256 threads / 32
#define ITERS  (BN / (CHUNKS * WAVES * 16))   // 16 iterations of 16 samples

__device__ __forceinline__ float fast_tanh(float x) {
#if __has_builtin(__builtin_amdgcn_tanhf)
  return __builtin_amdgcn_tanhf(x);               // V_TANH_F32 (gfx1250 trans op)
#elif __has_builtin(__builtin_amdgcn_tanh_f32)
  return __builtin_amdgcn_tanh_f32(x);
#else
  // stable tanh: 1 - 2/(e^{2x}+1); e->inf => 1, e->0 => -1 (no inf/inf)
  float e = __expf(2.0f * x);
  return 1.0f - 2.0f / (e + 1.0f);
#endif
}

__global__ __launch_bounds__(256) void subnet_fwd_d2_kernel(
    const float* __restrict__ x,   // [BN, SN]
    const float* __restrict__ W1,  // [SN, 1, NH1]
    const float* __restrict__ b1,  // [SN, NH1]
    const float* __restrict__ W2,  // [SN, NH1, NH2]
    const float* __restrict__ b2,  // [SN, NH2]
    const float* __restrict__ W3,  // [SN, NH2, 1]
    const float* __restrict__ b3,  // [SN]
    float* __restrict__ out,       // [BN, SN]  (raw, normalized later)
    float* __restrict__ sums)      // [3*SN]    {sum_o, sum_o2, sum_g2sq}
{
  const int s     = blockIdx.x;
  const int chunk = blockIdx.y;
  const int tid   = threadIdx.x;
  const int wave  = tid >> 5;
  const int lane  = tid & 31;
  const int m     = lane & 15;        // sample slot within 16-wide tile
  const bool hi   = (lane >= 16);     // upper half duplicates samples 0..15

  // ---- per-subnet parameters (uniform address -> scalar loads) ----
  float w1[NH1], bb1[NH1];
#pragma unroll
  for (int j = 0; j < NH1; ++j) { w1[j] = W1[s * NH1 + j]; bb1[j] = b1[s * NH1 + j]; }
  float bb2[NH2], w3[NH2];
#pragma unroll
  for (int n = 0; n < NH2; ++n) { bb2[n] = b2[s * NH2 + n]; w3[n] = W3[s * NH2 + n]; }
  const float bb3 = b3[s];

  // ---- A operand: W2^T, 16x12 tile (rows 6..15 & K 10..11 zero-padded) ----
  // A[M=n][K=k] = W2[k][n].  Per ISA layout: lanes 0-15 hold M=lane with
  // K = 4c+{0,1}; lanes 16-31 hold M=lane-16 with K = 4c+{2,3}.
  v2f A[3];
#pragma unroll
  for (int c = 0; c < 3; ++c) {
    const int k0 = 4 * c + (hi ? 2 : 0);
    const int k1 = k0 + 1;
    float a0 = 0.0f, a1 = 0.0f;
    if (m < NH2) {
      if (k0 < NH1) a0 = W2[s * NH1 * NH2 + k0 * NH2 + m];
      if (k1 < NH1) a1 = W2[s * NH1 * NH2 + k1 * NH2 + m];
    }
    A[c][0] = a0; A[c][1] = a1;
  }

  float acc_o = 0.0f, acc_o2 = 0.0f, acc_g = 0.0f;
  const int base = chunk * (WAVES * ITERS * 16) + wave * (ITERS * 16);

  for (int it = 0; it < ITERS; ++it) {
    const int b = base + it * 16 + m;
    const float xs = x[b * SN + s];

    // ---- layer 1 + analytic 1st/2nd derivatives wrt scalar input ----
    float h[12], dh[12], d2h[12];
#pragma unroll
    for (int j = 0; j < NH1; ++j) {
      const float u    = fmaf(xs, w1[j], bb1[j]);
      const float t    = fast_tanh(u);
      const float omt2 = fmaf(-t, t, 1.0f);        // 1 - t^2
      h[j]   = t;
      dh[j]  = omt2 * w1[j];                        // dh/dx
      d2h[j] = -2.0f * t * omt2 * w1[j] * w1[j];    // d2h/dx2
    }
    h[10] = h[11] = 0.0f; dh[10] = dh[11] = 0.0f; d2h[10] = d2h[11] = 0.0f;

    // ---- layer 2 matmuls on the matrix pipe: D[n][m] = sum_k W2[k][n]*X[k][m]
    v8f accV = {0,0,0,0,0,0,0,0};
    v8f accD = {0,0,0,0,0,0,0,0};
    v8f accH = {0,0,0,0,0,0,0,0};
#pragma unroll
    for (int c = 0; c < 3; ++c) {
      // B operand 4x16: VGPR0 = rows K=4c+0 (lanes 0-15) / 4c+2 (lanes 16-31),
      //                 VGPR1 = rows K=4c+1 / 4c+3.  One v_cndmask each.
      v2f Bv, Bd, Bh;
      Bv[0] = hi ? h[4 * c + 2]   : h[4 * c + 0];
      Bv[1] = hi ? h[4 * c + 3]   : h[4 * c + 1];
      Bd[0] = hi ? dh[4 * c + 2]  : dh[4 * c + 0];
      Bd[1] = hi ? dh[4 * c + 3]  : dh[4 * c + 1];
      Bh[0] = hi ? d2h[4 * c + 2] : d2h[4 * c + 0];
      Bh[1] = hi ? d2h[4 * c + 3] : d2h[4 * c + 1];
      accV = __builtin_amdgcn_wmma_f32_16x16x4_f32(false, A[c], false, Bv,
                                                   (short)0, accV, false, false);
      accD = __builtin_amdgcn_wmma_f32_16x16x4_f32(false, A[c], false, Bd,
                                                   (short)0, accD, false, false);
      accH = __builtin_amdgcn_wmma_f32_16x16x4_f32(false, A[c], false, Bh,
                                                   (short)0, accH, false, false);
    }

    // ---- epilogue: valid in lanes 0-15 (D VGPR n = unit n, lane = sample) ----
    float o = bb3, g = 0.0f;
#pragma unroll
    for (int n = 0; n < NH2; ++n) {
      const float vv   = accV[n] + bb2[n];
      const float z    = fast_tanh(vv);
      const float omz2 = fmaf(-z, z, 1.0f);
      const float dv   = accD[n];
      const float d2v  = accH[n];
      const float d2z  = omz2 * d2v - 2.0f * z * omz2 * dv * dv;
      o = fmaf(z,   w3[n], o);
      g = fmaf(d2z, w3[n], g);
    }
    if (!hi) {
      out[b * SN + s] = o;                 // final [B,S] layout; merged in L2
      acc_o  += o;
      acc_o2 += o * o;
      acc_g  += g * g;
    }
  }

  // ---- reduce partial sums: wave (hi lanes hold 0) -> LDS -> atomics ----
#pragma unroll
  for (int off = 16; off >= 1; off >>= 1) {
    acc_o  += __shfl_down(acc_o,  off, 32);
    acc_o2 += __shfl_down(acc_o2, off, 32);
    acc_g  += __shfl_down(acc_g,  off, 32);
  }
  __shared__ float red[3][WAVES];
  if (lane == 0) { red[0][wave] = acc_o; red[1][wave] = acc_o2; red[2][wave] = acc_g; }
  __syncthreads();
  if (tid == 0) {
    float so = 0.0f, so2 = 0.0f, sg = 0.0f;
#pragma unroll
    for (int w = 0; w < WAVES; ++w) { so += red[0][w]; so2 += red[1][w]; sg += red[2][w]; }
    atomicAdd(&sums[s],          so);    // global_atomic_add_f32
    atomicAdd(&sums[SN + s],     so2);
    atomicAdd(&sums[2 * SN + s], sg);
  }
}

__global__ __launch_bounds__(256) void stats_kernel(
    const float* __restrict__ sums, float* __restrict__ mean_o,
    float* __restrict__ rstd_o, float* __restrict__ loss_out)
{
  const int s = threadIdx.x;
  const float inv_b = 1.0f / (float)BN;
  const float mean  = sums[s] * inv_b;
  const float var   = fmaxf(fmaf(-mean, mean, sums[SN + s] * inv_b), 0.0f);
  const float sd    = fmaxf(sqrtf(var), 1e-10f);
  mean_o[s] = mean;
  rstd_o[s] = 1.0f / sd;
  float part = (sums[2 * SN + s] * inv_b) / sd;

  __shared__ float lred[SN];
  lred[s] = part;
  __syncthreads();
  for (int off = SN / 2; off > 0; off >>= 1) {
    if (s < off) lred[s] += lred[s + off];
    __syncthreads();
  }
  if (s == 0) loss_out[0] = 1e-3f * lred[0];
}

__global__ __launch_bounds__(256) void bn_normalize_kernel(
    float* __restrict__ out, const float* __restrict__ mean,
    const float* __restrict__ rstd)
{
  const int i = blockIdx.x * blockDim.x + threadIdx.x;   // float4 index
  float4 v = reinterpret_cast<float4*>(out)[i];
  const int s0 = (i * 4) & (SN - 1);                     // SN divides 4-groups
  v.x = (v.x - mean[s0 + 0]) * rstd[s0 + 0];
  v.y = (v.y - mean[s0 + 1]) * rstd[s0 + 1];
  v.z = (v.z - mean[s0 + 2]) * rstd[s0 + 2];
  v.w = (v.w - mean[s0 + 3]) * rstd[s0 + 3];
  reinterpret_cast<float4*>(out)[i] = v;
}

extern "C" void kernel_launch(void* const* d_in, const int* in_sizes, int n_in,
                              void* d_out, int out_size, void* d_ws, size_t ws_size,
                              hipStream_t stream) {
  const float* x  = (const float*)d_in[0];
  const float* W1 = (const float*)d_in[1];
  const float* b1 = (const float*)d_in[2];
  const float* W2 = (const float*)d_in[3];
  const float* b2 = (const float*)d_in[4];
  const float* W3 = (const float*)d_in[5];
  const float* b3 = (const float*)d_in[6];

  float* out  = (float*)d_out;                 // [BN*SN] + 1 loss scalar
  float* sums = (float*)d_ws;                  // 3*SN partial sums
  float* mean = sums + 3 * SN;
  float* rstd = sums + 4 * SN;

  hipMemsetAsync(d_ws, 0, 3 * SN * sizeof(float), stream);  // capture-safe

  dim3 g1(SN, CHUNKS);
  subnet_fwd_d2_kernel<<<g1, 256, 0, stream>>>(x, W1, b1, W2, b2, W3, b3, out, sums);
  stats_kernel<<<1, SN, 0, stream>>>(sums, mean, rstd, out + (size_t)BN * SN);
  bn_normalize_kernel<<<(BN * SN / 4) / 256, 256, 0, stream>>>(out, mean, rstd);
}